// VGAE_84129819394641
// MI455X (gfx1250) — compile-verified
//
#include <hip/hip_runtime.h>
#include <hip/hip_bf16.h>

typedef float v2f __attribute__((ext_vector_type(2)));
typedef float v8f __attribute__((ext_vector_type(8)));

// ---------------------------------------------------------------------------
// Hardware f32 atomic add (lowers to non-returning global_atomic_add_f32)
// ---------------------------------------------------------------------------
__device__ __forceinline__ void atomic_add_f32(float* p, float v) {
    __hip_atomic_fetch_add(p, v, __ATOMIC_RELAXED, __HIP_MEMORY_SCOPE_AGENT);
}

// ---------------------------------------------------------------------------
// Utility kernels
// ---------------------------------------------------------------------------
__global__ void zero_f32(float* __restrict__ p, long long n) {
    long long i = (long long)blockIdx.x * blockDim.x + threadIdx.x;
    if (i < n) p[i] = 0.0f;
}

__global__ void deg_accum(const int* __restrict__ dst, float* __restrict__ deg, int E) {
    int e = blockIdx.x * blockDim.x + threadIdx.x;
    if (e < E) atomic_add_f32(&deg[dst[e]], 1.0f);
}

__global__ void dinv_kernel(float* __restrict__ deg, int N) {
    int i = blockIdx.x * blockDim.x + threadIdx.x;
    if (i < N) deg[i] = rsqrtf(deg[i] + 1.0f);  // self-loop: deg + 1
}

// ---------------------------------------------------------------------------
// Dense fp32 GEMM via V_WMMA_F32_16X16X4_F32.
//
//  - 256-thread block = 8 waves; each wave owns a 16-row strip, the block
//    covers 128 rows. The B column panel (K x 32) is staged in LDS once per
//    block and shared by all 8 waves (<= 32 KB; WGP LDS is 320 KB).
//  - Each wave computes a 16x32 tile: two independent v8f accumulators
//    (two WMMA chains pipeline instead of stalling on the RAW hazard) and
//    A is loaded once per K-step as a float2 (global_load_b64).
//
// VGPR layouts per CDNA5 ISA 7.12.2:
//   A (16x4, MxK):  lanes 0-15 -> M=lane, {v0,v1}={K0,K1}; lanes 16-31 {K2,K3}
//   B (4x16, KxN):  lanes 0-15 -> N=lane, {v0,v1}={K0,K1}; lanes 16-31 {K2,K3}
//   C/D (16x16):    vR: lanes 0-15 -> M=R,   lanes 16-31 -> M=R+8; N=lane&15
//
// Requires: K % 4 == 0, K <= 256, Ncol % 32 == 0, rows handled with a
// wave-uniform tail guard (M=100000 is a multiple of 16).
// ---------------------------------------------------------------------------
#define GEMM_TN 32
#define GEMM_KMAX 256

__global__ __launch_bounds__(256)
void wmma_gemm_f32(const float* __restrict__ A, const float* __restrict__ B,
                   float* __restrict__ C, int M, int K, int Ncol) {
    __shared__ float Bs[GEMM_KMAX * GEMM_TN];

    const int tid = threadIdx.x;
    const int tn  = blockIdx.y * GEMM_TN;

    // Cooperative stage of B[K x 32] -> LDS (coalesced: consecutive tid ->
    // consecutive columns within a row segment).
    for (int idx = tid; idx < K * GEMM_TN; idx += 256) {
        int k = idx / GEMM_TN;
        int n = idx % GEMM_TN;
        Bs[idx] = B[(long long)k * Ncol + tn + n];
    }
    __syncthreads();

    const int wave = tid >> 5;
    const int lane = tid & 31;
    const int tm   = (blockIdx.x * 8 + wave) * 16;   // uniform per wave
    if (tm >= M) return;                             // whole-wave exit (post-barrier)

    const int half = lane >> 4;                      // 0: K{0,1}, 1: K{2,3}
    const int l15  = lane & 15;
    const float2* __restrict__ arow =
        (const float2*)(A + (long long)(tm + l15) * K);

    v8f acc0 = {};
    v8f acc1 = {};
    for (int k = 0; k < K; k += 4) {
        int ka = k + half * 2;                       // even
        float2 av = arow[ka >> 1];                   // global_load_b64
        v2f a;  a.x = av.x;  a.y = av.y;
        v2f b0, b1;
        b0.x = Bs[ka * GEMM_TN + l15];
        b0.y = Bs[(ka + 1) * GEMM_TN + l15];
        b1.x = Bs[ka * GEMM_TN + 16 + l15];
        b1.y = Bs[(ka + 1) * GEMM_TN + 16 + l15];
        acc0 = __builtin_amdgcn_wmma_f32_16x16x4_f32(
            false, a, false, b0, (short)0, acc0, false, false);
        acc1 = __builtin_amdgcn_wmma_f32_16x16x4_f32(
            false, a, false, b1, (short)0, acc1, false, false);
    }
#pragma unroll
    for (int r = 0; r < 8; ++r) {
        int row = tm + r + half * 8;
        C[(long long)row * Ncol + tn + l15]      = acc0[r];
        C[(long long)row * Ncol + tn + 16 + l15] = acc1[r];
    }
}

// ---------------------------------------------------------------------------
// GCN layer 1 (C=128): init self term + bias, wave-per-edge scatter, ReLU
// ---------------------------------------------------------------------------
__global__ void init_agg1(const float* __restrict__ hpre, const float* __restrict__ dinv,
                          const float* __restrict__ b1, float* __restrict__ agg,
                          int N, int C) {
    long long i = (long long)blockIdx.x * blockDim.x + threadIdx.x;
    long long total = (long long)N * C;
    if (i < total) {
        int node = (int)(i / C);
        int c    = (int)(i % C);
        float di = dinv[node];
        agg[i] = hpre[i] * di * di + b1[c];
    }
}

__global__ __launch_bounds__(256)
void edge_agg128(const int* __restrict__ src, const int* __restrict__ dst,
                 const float* __restrict__ dinv, const float* __restrict__ hpre,
                 float* __restrict__ agg, int E) {
    int e = blockIdx.x * (blockDim.x >> 5) + (threadIdx.x >> 5);
    if (e >= E) return;
    int lane = threadIdx.x & 31;
    int s = src[e], d = dst[e];
    float coef = dinv[s] * dinv[d];
    const float4* __restrict__ hp = (const float4*)(hpre + (long long)s * 128);
    float4 v = hp[lane];                               // 4 channels / lane
    float* ad = agg + (long long)d * 128 + lane * 4;
    atomic_add_f32(ad + 0, coef * v.x);
    atomic_add_f32(ad + 1, coef * v.y);
    atomic_add_f32(ad + 2, coef * v.z);
    atomic_add_f32(ad + 3, coef * v.w);
}

__global__ void relu_kernel(float* __restrict__ p, long long n) {
    long long i = (long long)blockIdx.x * blockDim.x + threadIdx.x;
    if (i < n) p[i] = fmaxf(p[i], 0.0f);
}

// ---------------------------------------------------------------------------
// GCN layers 2/3 (C=64 each, mu + logvar together)
// ---------------------------------------------------------------------------
__global__ void init_agg2(const float* __restrict__ gmu, const float* __restrict__ glv,
                          const float* __restrict__ dinv,
                          const float* __restrict__ bmu, const float* __restrict__ blv,
                          float* __restrict__ mu, float* __restrict__ lv,
                          int N, int C) {
    long long i = (long long)blockIdx.x * blockDim.x + threadIdx.x;
    long long total = (long long)N * C;
    if (i < total) {
        int node = (int)(i / C);
        int c    = (int)(i % C);
        float di = dinv[node];
        float d2 = di * di;
        mu[i] = gmu[i] * d2 + bmu[c];
        lv[i] = glv[i] * d2 + blv[c];
    }
}

__global__ __launch_bounds__(256)
void edge_agg64x2(const int* __restrict__ src, const int* __restrict__ dst,
                  const float* __restrict__ dinv,
                  const float* __restrict__ gmu, const float* __restrict__ glv,
                  float* __restrict__ mu, float* __restrict__ lv, int E) {
    int e = blockIdx.x * (blockDim.x >> 5) + (threadIdx.x >> 5);
    if (e >= E) return;
    int lane = threadIdx.x & 31;
    int s = src[e], d = dst[e];
    float coef = dinv[s] * dinv[d];
    const float2* __restrict__ ms = (const float2*)(gmu + (long long)s * 64);
    const float2* __restrict__ ls = (const float2*)(glv + (long long)s * 64);
    float2 vm = ms[lane];                              // 2 mu channels / lane
    float2 vl = ls[lane];                              // 2 lv channels / lane
    float* md = mu + (long long)d * 64 + lane * 2;
    float* ld = lv + (long long)d * 64 + lane * 2;
    atomic_add_f32(md + 0, coef * vm.x);
    atomic_add_f32(md + 1, coef * vm.y);
    atomic_add_f32(ld + 0, coef * vl.x);
    atomic_add_f32(ld + 1, coef * vl.y);
}

// ---------------------------------------------------------------------------
// z = mu + eps * exp(0.5 * logvar)
// ---------------------------------------------------------------------------
__global__ void z_kernel(const float* __restrict__ mu, const float* __restrict__ lv,
                         const float* __restrict__ eps, float* __restrict__ z,
                         long long n) {
    long long i = (long long)blockIdx.x * blockDim.x + threadIdx.x;
    if (i < n) z[i] = mu[i] + eps[i] * expf(0.5f * lv[i]);
}

// ---------------------------------------------------------------------------
// Decoder: sigmoid(<z_src, z_dst>) per edge; wave32 per edge, 2 dims/lane
// ---------------------------------------------------------------------------
__global__ __launch_bounds__(256)
void decoder_kernel(const int* __restrict__ src, const int* __restrict__ dst,
                    const float* __restrict__ z, float* __restrict__ dec, int E) {
    int e = blockIdx.x * (blockDim.x >> 5) + (threadIdx.x >> 5);
    if (e >= E) return;
    int lane = threadIdx.x & 31;
    int s = src[e], d = dst[e];
    const float2* __restrict__ zs = (const float2*)(z + (long long)s * 64);
    const float2* __restrict__ zd = (const float2*)(z + (long long)d * 64);
    float2 a = zs[lane];
    float2 b = zd[lane];
    float p = a.x * b.x + a.y * b.y;
#pragma unroll
    for (int off = 16; off > 0; off >>= 1)
        p += __shfl_xor(p, off, 32);
    if (lane == 0) dec[e] = 1.0f / (1.0f + expf(-p));
}

// ---------------------------------------------------------------------------
// Host-side launcher
// ---------------------------------------------------------------------------
extern "C" void kernel_launch(void* const* d_in, const int* in_sizes, int n_in,
                              void* d_out, int out_size, void* d_ws, size_t ws_size,
                              hipStream_t stream) {
    const float* x    = (const float*)d_in[0];
    const int*   edge = (const int*)d_in[1];
    const float* eps  = (const float*)d_in[2];
    const float* W1   = (const float*)d_in[3];
    const float* b1   = (const float*)d_in[4];
    const float* Wmu  = (const float*)d_in[5];
    const float* bmu  = (const float*)d_in[6];
    const float* Wlv  = (const float*)d_in[7];
    const float* blv  = (const float*)d_in[8];

    const int E     = in_sizes[1] / 2;          // 1,600,000
    const int C_h   = in_sizes[4];              // 128
    const int C_out = in_sizes[6];              // 64
    const int N     = in_sizes[2] / C_out;      // 100,000
    const int C_in  = in_sizes[0] / N;          // 256

    const int* src = edge;
    const int* dst = edge + E;

    // Workspace layout (floats): dinv[N] | hpre[N*C_h] | h[N*C_h]
    // g_mu / g_lv overlay hpre (dead after layer-1 scatter): 2*N*C_out == N*C_h
    float* ws   = (float*)d_ws;
    float* dinv = ws;
    float* hpre = ws + N;
    float* h    = hpre + (long long)N * C_h;
    float* gmu  = hpre;
    float* glv  = hpre + (long long)N * C_out;

    // Output layout: z[N*C_out] | mu[N*C_out] | logvar[N*C_out] | dec[E]
    float* z   = (float*)d_out;
    float* mu  = z  + (long long)N * C_out;
    float* lv  = mu + (long long)N * C_out;
    float* dec = lv + (long long)N * C_out;

    const int tpb = 256;
    const int wpb = tpb / 32;   // waves per block = 8

    // 1) degree + dinv
    zero_f32<<<(N + tpb - 1) / tpb, tpb, 0, stream>>>(dinv, N);
    deg_accum<<<(E + tpb - 1) / tpb, tpb, 0, stream>>>(dst, dinv, E);
    dinv_kernel<<<(N + tpb - 1) / tpb, tpb, 0, stream>>>(dinv, N);

    // 2) h_pre = x @ W1  (WMMA f32, LDS-staged B panel)
    {
        int mtiles = (N + 15) / 16;
        dim3 grid((mtiles + 7) / 8, C_h / GEMM_TN);
        wmma_gemm_f32<<<grid, 256, 0, stream>>>(x, W1, hpre, N, C_in, C_h);
    }

    // 3) layer-1 aggregation + ReLU -> h
    long long nh = (long long)N * C_h;
    init_agg1<<<(int)((nh + tpb - 1) / tpb), tpb, 0, stream>>>(hpre, dinv, b1, h, N, C_h);
    edge_agg128<<<(E + wpb - 1) / wpb, tpb, 0, stream>>>(src, dst, dinv, hpre, h, E);
    relu_kernel<<<(int)((nh + tpb - 1) / tpb), tpb, 0, stream>>>(h, nh);

    // 4) g_mu = h @ Wmu ; g_lv = h @ Wlv  (WMMA f32)
    {
        int mtiles = (N + 15) / 16;
        dim3 grid((mtiles + 7) / 8, C_out / GEMM_TN);
        wmma_gemm_f32<<<grid, 256, 0, stream>>>(h, Wmu, gmu, N, C_h, C_out);
        wmma_gemm_f32<<<grid, 256, 0, stream>>>(h, Wlv, glv, N, C_h, C_out);
    }

    // 5) layer-2/3 aggregation -> mu, logvar (directly in d_out)
    long long nz = (long long)N * C_out;
    init_agg2<<<(int)((nz + tpb - 1) / tpb), tpb, 0, stream>>>(gmu, glv, dinv, bmu, blv,
                                                              mu, lv, N, C_out);
    edge_agg64x2<<<(E + wpb - 1) / wpb, tpb, 0, stream>>>(src, dst, dinv, gmu, glv,
                                                          mu, lv, E);

    // 6) reparameterization
    z_kernel<<<(int)((nz + tpb - 1) / tpb), tpb, 0, stream>>>(mu, lv, eps, z, nz);

    // 7) inner-product decoder
    decoder_kernel<<<(E + wpb - 1) / wpb, tpb, 0, stream>>>(src, dst, z, dec, E);
}